// SpectralConv1d_39548058862298
// MI455X (gfx1250) — compile-verified
//
#include <hip/hip_runtime.h>
#include <hip/hip_bf16.h>
#include <stddef.h>

typedef float v2f __attribute__((ext_vector_type(2)));
typedef float v8f __attribute__((ext_vector_type(8)));
typedef unsigned int v4u __attribute__((ext_vector_type(4)));
typedef int v4i __attribute__((ext_vector_type(4)));
typedef int v8i __attribute__((ext_vector_type(8)));

#define MODES 2048
#define CIN   128
#define COUT  128
#define BATCH 16
#define NLEN  8192
#define LOUT  4097   // N/2 + 1

// ---------------------------------------------------------------------------
// Kernel 1: forward RFHT (length 8192, power of two) per (b, cin) row.
// out = concat(E + cos*O, -sin*O) recursion == bit-reversal + DIT butterflies.
// Only the first 2048 bins are kept (all the mixing needs).
// ---------------------------------------------------------------------------
__global__ __launch_bounds__(256)
void fht_fwd_kernel(const float* __restrict__ x, float* __restrict__ xh) {
    __shared__ float y[NLEN];                       // 32 KB
    const int row = blockIdx.x;                     // 0..2047 = b*128 + cin
    const int tid = threadIdx.x;
    const float* xr = x + (size_t)row * NLEN;

    for (int j = tid; j < NLEN; j += 256)
        y[__brev((unsigned)j) >> 19] = xr[j];       // 13-bit reverse scatter

    for (int st = 1; st <= 13; ++st) {
        __syncthreads();
        const int half = 1 << (st - 1);
        const int n    = half << 1;
        for (int t = tid; t < NLEN / 2; t += 256) {
            const int k    = t & (half - 1);
            const int base = ((t >> (st - 1)) << st) + k;
            const float E = y[base];
            const float O = y[base + half];
            float sn, cs;
            __sincosf(6.283185307179586f * (float)k / (float)n, &sn, &cs);
            y[base]        = E + cs * O;            // "re" half
            y[base + half] = -sn * O;               // "im" half
        }
    }
    __syncthreads();
    float* dst = xh + (size_t)row * MODES;
    for (int j = tid; j < MODES; j += 256) dst[j] = y[j];
}

// ---------------------------------------------------------------------------
// Kernel 2: Hartley channel mixing with WMMA f32 16x16x4 + TDM weight DMA.
//   P[b,o,m] = sum_i (x1+x1n)[b,i,m] * W[i,o,m]
//   Q[b,o,m] = sum_i (x1-x1n)[b,i,m] * W[i,o,m]
// (mixed = 0.5*(P[m] - Q[mirror(m)]) is folded into kernel 3's load.)
// Block = 8 modes x all 128 outputs, 512 threads (16 waves).
// Weights are moved by TENSOR_LOAD_TO_LDS as a 3-D tile:
//   dim0 = 8 modes (contiguous), dim1 = 128 outs (stride 2048),
//   dim2 = 8 cins (stride 128*2048), with pad_enable inserting 1 DWORD per 8
//   so each (i,o) row lands at stride 9 floats (bank-conflict-free fetches).
// Double-buffered: TDM + s/d staging for chunk c+1 overlap chunk c's WMMAs.
// ---------------------------------------------------------------------------
#define MT   8                   // modes per block
#define IC   8                   // Cin chunk per LDS stage
#define WROW 9                   // padded m-row stride in floats (8 + 1 pad)
#define WBUF (IC * COUT * WROW)  // 9216 floats per weight buffer (36 KB)
#define IP   12                  // padded i-stride for s/d (bank spread)

__device__ __forceinline__ v8f wmma_f32(v2f a, v2f b, v8f c) {
    return __builtin_amdgcn_wmma_f32_16x16x4_f32(
        false, a, false, b, (short)0, c, false, false);
}

// Build the Tensor DMA Descriptor (D#) and issue TENSOR_LOAD_TO_LDS.
// This toolchain's builtin takes 6 args: (g0, g1, g2, g3, g4, cpol).
__device__ __forceinline__ void tdm_load_weights(const float* wgt,
                                                 unsigned lds_bytes,
                                                 int i0, int m0) {
    const unsigned long long ga = (unsigned long long)(size_t)wgt
        + 4ull * ((unsigned long long)i0 * COUT * MODES + (unsigned long long)m0);
    v4u g0;
    g0.x = 1u;                                      // count=1, user descriptor
    g0.y = lds_bytes;                               // lds_addr (bytes)
    g0.z = (unsigned)(ga & 0xFFFFFFFFu);            // global_addr[31:0]
    g0.w = (unsigned)((ga >> 32) & 0x01FFFFFFu)     // global_addr[56:32]
         | 0x80000000u;                             // type=2 ("image")
    v8i g1;
    g1[0] = (2 << 16)                               // data_size = 4 bytes
          | (1 << 20)                               // pad_enable
          | (2 << 22)                               // pad_interval: every 8 DWORDs
          | (0 << 25);                              // pad_amount: 1 DWORD
    g1[1] = (int)(2048u << 16);                     // tensor_dim0[15:0] = 2048
    g1[2] = (int)(128u << 16);                      // dim0 hi=0 | tensor_dim1 lo = 128
    g1[3] = (int)(8u << 16);                        // dim1 hi=0 | tile_dim0 = 8 (m)
    g1[4] = (int)(128u | (8u << 16));               // tile_dim1 = 128 (o), tile_dim2 = 8 (i)
    g1[5] = (int)2048;                              // tensor_dim0_stride[31:0] = MODES
    g1[6] = (int)((262144u & 0xFFFFu) << 16);       // stride0 hi=0 | stride1 lo16
    g1[7] = (int)(262144u >> 16);                   // tensor_dim1_stride[47:16] (COUT*MODES)
    v4i g2;
    g2[0] = 128;                                    // tensor_dim2 >= tile_dim2
    g2[1] = 0;                                      // tensor_dim3 (unused)
    g2[2] = 0;                                      // tensor_dim2_stride (tile_dim3=0)
    g2[3] = 0;                                      // tile_dim3 = 0 (unused)
    v4i g3 = {0, 0, 0, 0};                          // no dim4
    v8i g4 = {0, 0, 0, 0, 0, 0, 0, 0};              // unused trailing group
    __builtin_amdgcn_tensor_load_to_lds(g0, g1, g2, g3, g4, 0);
}

__device__ __forceinline__ void stage_sd(const float* __restrict__ xh, int tid,
                                         int m0, int i0,
                                         float* __restrict__ sbuf,
                                         float* __restrict__ dbuf) {
    for (int q = tid; q < MT * BATCH * IC; q += 512) {
        const int i = q & 7;
        const int b = (q >> 3) & 15;
        const int m = q >> 7;                       // 0..7
        const int mm  = m0 + m;
        const int mir = (MODES - mm) & (MODES - 1);
        const size_t base = ((size_t)b * CIN + i0 + i) * MODES;
        const float xp = xh[base + mm];
        const float xn = xh[base + mir];
        sbuf[(m * BATCH + b) * IP + i] = xp + xn;
        dbuf[(m * BATCH + b) * IP + i] = xp - xn;
    }
}

__global__ __launch_bounds__(512)
void mix_wmma_kernel(const float* __restrict__ xh, const float* __restrict__ wgt,
                     float* __restrict__ P, float* __restrict__ Q) {
    __shared__ float w_lds[2][WBUF];                // 72 KB : [i][o][m(pad9)]
    __shared__ float s_lds[2][MT * BATCH * IP];     // 12 KB : [m][b][i(pad12)]
    __shared__ float d_lds[2][MT * BATCH * IP];     // 12 KB

    const int tid   = threadIdx.x;
    const int ln    = tid & 31;
    const int wv    = tid >> 5;
    const int m_loc = wv & 7;
    const int ohalf = wv >> 3;                      // 0/1
    const int m0    = blockIdx.x * MT;
    const int bl    = ln & 15;                      // A row (=batch) / B col (=o)
    const int hi    = ln >> 4;                      // K-half select per ISA layout

    const unsigned wbase = (unsigned)(size_t)&w_lds[0][0];

    v8f Pacc[4] = {};
    v8f Qacc[4] = {};

    // ---- prologue: chunk 0 in flight ----
    if (wv == 0) tdm_load_weights(wgt, wbase, 0, m0);
    stage_sd(xh, tid, m0, 0, s_lds[0], d_lds[0]);
    if (wv == 0) __builtin_amdgcn_s_wait_tensorcnt(0);
    __syncthreads();

    for (int c = 0; c < CIN / IC; ++c) {
        const int cur  = c & 1;
        const int nxt  = cur ^ 1;
        const bool more = (c + 1) < (CIN / IC);
        if (more) {                                  // overlap next chunk's DMA
            if (wv == 0)
                tdm_load_weights(wgt, wbase + (unsigned)(nxt * WBUF * 4),
                                 (c + 1) * IC, m0);
            stage_sd(xh, tid, m0, (c + 1) * IC, s_lds[nxt], d_lds[nxt]);
        }
        const float* wl = w_lds[cur];
        const float* sl = s_lds[cur];
        const float* dl = d_lds[cur];
        for (int istep = 0; istep < 2; ++istep) {
            const int ib = istep * 4 + hi * 2;       // K base for this lane-half
            v2f a_s, a_d;
            a_s.x = sl[(m_loc * BATCH + bl) * IP + ib];
            a_s.y = sl[(m_loc * BATCH + bl) * IP + ib + 1];
            a_d.x = dl[(m_loc * BATCH + bl) * IP + ib];
            a_d.y = dl[(m_loc * BATCH + bl) * IP + ib + 1];
            for (int ot = 0; ot < 4; ++ot) {
                const int o = (ohalf * 4 + ot) * 16 + bl;
                v2f bm;
                bm.x = wl[(ib * COUT + o) * WROW + m_loc];
                bm.y = wl[((ib + 1) * COUT + o) * WROW + m_loc];
                Pacc[ot] = wmma_f32(a_s, bm, Pacc[ot]);
                Qacc[ot] = wmma_f32(a_d, bm, Qacc[ot]);
            }
        }
        if (more) {
            if (wv == 0) __builtin_amdgcn_s_wait_tensorcnt(0);
            __syncthreads();
        }
    }

    // ---- write back D tiles (VGPR r: rows r / r+8 per lane-half) ----
    const int m = m0 + m_loc;
    for (int ot = 0; ot < 4; ++ot) {
        const int o = (ohalf * 4 + ot) * 16 + bl;
        for (int r = 0; r < 8; ++r) {
            const int b = r + hi * 8;
            const size_t idx = ((size_t)b * COUT + o) * MODES + m;
            P[idx] = Pacc[ot][r];
            Q[idx] = Qacc[ot][r];
        }
    }
}

// ---------------------------------------------------------------------------
// Kernel 3: inverse RFHT of length 4097 (odd, recursive pad-and-truncate),
// realized as an 8192-leaf bottom-up butterfly with per-level sizes
// L = {4097,2049,...,3,2,1}: leaf bitrev13(p) holds spectrum[p] (pads = 0);
// each level combines children of length Lc (twiddle pi*k/Lc) and keeps the
// first Lp = parent-length elements (the odd-length truncation).
// Spectrum[p] = 0.5*(P[p] - Q[mirror(p)]) for p<2048, else 0. Output /4097.
// ---------------------------------------------------------------------------
__global__ __launch_bounds__(256)
void fht_inv_kernel(const float* __restrict__ P, const float* __restrict__ Q,
                    float* __restrict__ out) {
    __shared__ float bufA[NLEN];                    // 32 KB
    __shared__ float bufB[NLEN];                    // 32 KB
    const int row = blockIdx.x;                     // 0..2047 = b*128 + cout
    const int tid = threadIdx.x;

    // combined spectrum into bufB[0:2048]
    for (int p = tid; p < MODES; p += 256) {
        const float pv = P[(size_t)row * MODES + p];
        const float qv = Q[(size_t)row * MODES + ((MODES - p) & (MODES - 1))];
        bufB[p] = 0.5f * (pv - qv);
    }
    __syncthreads();
    // scatter to leaves
    for (int j = tid; j < NLEN; j += 256) {
        const int p = __brev((unsigned)j) >> 19;
        bufA[j] = (p < MODES) ? bufB[p] : 0.0f;
    }
    float* cur = bufA;
    float* nxt = bufB;
    const int Ltab[14] = {4097, 2049, 1025, 513, 257, 129, 65, 33, 17, 9, 5, 3, 2, 1};
#pragma unroll
    for (int cl = 13; cl >= 1; --cl) {
        __syncthreads();
        const int Lc = Ltab[cl];
        const int Lp = Ltab[cl - 1];
        const int Sc = NLEN >> cl;                  // child slot stride
        const int Sp = Sc << 1;                     // parent slot stride
        const int total = (1 << (cl - 1)) * Lc;
        for (int t = tid; t < total; t += 256) {
            const int s = t / Lc;                   // parent node (const divisor)
            const int k = t - s * Lc;
            const int base = s * Sp;
            const float E = cur[base + k];
            const float O = cur[base + Sc + k];
            float sn, cs;
            __sincosf(3.14159265358979f * (float)k / (float)Lc, &sn, &cs);
            nxt[base + k] = E + cs * O;
            const int ip = Lc + k;
            if (ip < Lp) nxt[base + ip] = -sn * O;  // odd-length truncation
        }
        float* tmp = cur; cur = nxt; nxt = tmp;
    }
    __syncthreads();
    for (int p = tid; p < LOUT; p += 256)
        out[(size_t)row * LOUT + p] = cur[p] * (1.0f / (float)LOUT);
}

// ---------------------------------------------------------------------------
extern "C" void kernel_launch(void* const* d_in, const int* in_sizes, int n_in,
                              void* d_out, int out_size, void* d_ws, size_t ws_size,
                              hipStream_t stream) {
    (void)in_sizes; (void)n_in; (void)out_size;
    const float* x = (const float*)d_in[0];                 // [16][128][8192]
    const float* w = (const float*)d_in[1];                 // [128][128][2048]
    float* out = (float*)d_out;                             // [16][128][4097]

    const size_t nrow = (size_t)BATCH * CIN;                // 2048
    float* xh = (float*)d_ws;                               // 16 MB
    float* Pb = xh + nrow * MODES;                          // 16 MB
    float* Qb = Pb + nrow * MODES;                          // 16 MB
    if (ws_size < 3u * nrow * MODES * sizeof(float)) return;

    fht_fwd_kernel<<<dim3((unsigned)nrow), dim3(256), 0, stream>>>(x, xh);
    mix_wmma_kernel<<<dim3(MODES / MT), dim3(512), 0, stream>>>(xh, w, Pb, Qb);
    fht_inv_kernel<<<dim3((unsigned)nrow), dim3(256), 0, stream>>>(Pb, Qb, out);
}